// DiagonalS5SSM_63015760167063
// MI455X (gfx1250) — compile-verified
//
#include <hip/hip_runtime.h>
#include <hip/hip_bf16.h>
#include <stdint.h>

#define D_MODEL 512
#define D_STATE 256
#define BATCH   16
#define SEQ     2048
#define M_TOTAL (BATCH * SEQ)   // 32768
#define LDSP    516             // padded LDS row (dwords): 16B-aligned rows, conflict-free frag reads

typedef float v2f __attribute__((ext_vector_type(2)));
typedef float v8f __attribute__((ext_vector_type(8)));

// ---------------- workspace layout (in floats) ----------------
// W1p  [512*512]   : pair-interleaved (k>>1)*1024 + j*2 + (k&1); j<256 re, j>=256 im of bbar^T
// W2p  [1024*512]  : pair-interleaved (k>>1)*1024 + d*2 + (k&1); k rows: [c_re^T | -c_im^T | D^T]
// abar [512]       : re[256], im[256]
// buf  [32768][512]: row m=b*SEQ+l : [states_re 0..255 | states_im 0..255]
#define WS_W1   0
#define WS_W2   (512 * 512)
#define WS_ABAR (WS_W2 + 1024 * 512)
#define WS_BUF  (WS_ABAR + 512)

// ---------------- WMMA f32 16x16x4 (ISA 7.12.2 layouts) ----------------
__device__ __forceinline__ v8f wmma_f32(v2f a, v2f b, v8f c) {
  return __builtin_amdgcn_wmma_f32_16x16x4_f32(
      /*neg_a=*/false, a, /*neg_b=*/false, b,
      /*c_mod=*/(short)0, c, /*reuse_a=*/false, /*reuse_b=*/false);
}

// B fragment from pair-interleaved weights: one b64 load gives {W[kb][col], W[kb+1][col]}
__device__ __forceinline__ v2f load_b_pair(const float* __restrict__ Wp, int k,
                                           int n0, int lane) {
  int col = n0 + (lane & 15);
  int kb  = k + ((lane >> 4) << 1);  // even
  return *(const v2f*)(Wp + (size_t)(kb >> 1) * 1024 + col * 2);
}

// A fragment from padded LDS tile (row-major 16 x 512, pitch LDSP)
__device__ __forceinline__ v2f load_a_lds(const float* As, int k, int lane) {
  int row = lane & 15;
  int kb  = k + ((lane >> 4) << 1);
  return *(const v2f*)(As + row * LDSP + kb);
}

// C/D store: lane holds col (lane&15); VGPR i -> row i + 8*(lane>=16)
__device__ __forceinline__ void store_c(float* __restrict__ Y, int ldc, int m0,
                                        int n0, int lane, v8f c) {
  int col = n0 + (lane & 15);
  int mb  = m0 + ((lane >> 4) << 3);
#pragma unroll
  for (int i = 0; i < 8; ++i) Y[(size_t)(mb + i) * ldc + col] = c[i];
}

// Async-stage a 16x512 f32 tile (row-major, row stride 512) into padded LDS.
// 2048 16B chunks; 256 threads x 8 chunks, global reads fully coalesced.
__device__ __forceinline__ void stage_tile_async(const float* __restrict__ g0,
                                                 float* lds_tile, int tid) {
#pragma unroll
  for (int i = 0; i < 8; ++i) {
    int c   = tid + i * 256;
    int row = c >> 7;
    int kq  = (c & 127) << 2;
    uint32_t lds = (uint32_t)(uintptr_t)(&lds_tile[row * LDSP + kq]);
    unsigned long long g = (unsigned long long)(uintptr_t)(g0 + row * 512 + kq);
    asm volatile("global_load_async_to_lds_b128 %0, %1, off"
                 :: "v"(lds), "v"(g) : "memory");
  }
}

__device__ __forceinline__ void wait_async_all() {
  asm volatile("s_wait_asynccnt 0x0" ::: "memory");
}

// ---------------- setup: discretization -> fused W1p + abar ----------------
__global__ __launch_bounds__(256) void setup_w1_kernel(
    const float* __restrict__ lrl, const float* __restrict__ limg,
    const float* __restrict__ logdt, const float* __restrict__ Bre,
    const float* __restrict__ Bim, float* __restrict__ ws) {
  int tid = blockIdx.x * blockDim.x + threadIdx.x;  // 512*256 threads
  int n = tid & 255;
  int d = tid >> 8;
  if (d >= D_MODEL) return;

  float dt  = log1pf(expf(logdt[0])) + 1.0e-4f;  // softplus + eps
  float lre = -expf(lrl[n]);
  float lim = limg[n];
  float er  = expf(dt * lre);
  float are = er * cosf(dt * lim);
  float aim = er * sinf(dt * lim);
  // f = (abar - 1) / lambda  (complex)
  float inv = 1.0f / (lre * lre + lim * lim);
  float xr  = are - 1.0f;
  float fre = (xr * lre + aim * lim) * inv;
  float fim = (aim * lre - xr * lim) * inv;
  // bbar[n][d] = f * (Bre + i*Bim)
  float br = Bre[n * D_MODEL + d];
  float bi = Bim[n * D_MODEL + d];
  float* W1 = ws + WS_W1;
  size_t base = (size_t)(d >> 1) * 1024 + (d & 1);
  W1[base + (size_t)n * 2]         = fre * br - fim * bi;  // col j=n       (re)
  W1[base + (size_t)(n + 256) * 2] = fre * bi + fim * br;  // col j=n+256   (im)
  if (d == 0) {
    float* ab = ws + WS_ABAR;
    ab[n] = are;
    ab[256 + n] = aim;
  }
}

// ---------------- setup: fused output weight W2p ----------------
__global__ __launch_bounds__(256) void setup_w2_kernel(
    const float* __restrict__ Cre, const float* __restrict__ Cim,
    const float* __restrict__ Dw, float* __restrict__ ws) {
  int tid = blockIdx.x * blockDim.x + threadIdx.x;  // 1024*512 threads
  int d = tid & 511;
  int k = tid >> 9;
  if (k >= 1024) return;
  float v;
  if (k < 256)      v =  Cre[d * D_STATE + k];          // c_re^T
  else if (k < 512) v = -Cim[d * D_STATE + (k - 256)];  // -c_im^T
  else              v =  Dw[d * D_MODEL + (k - 512)];   // D^T
  ws[WS_W2 + (size_t)(k >> 1) * 1024 + (size_t)d * 2 + (k & 1)] = v;
}

// ---------------- GEMM1: buf = x @ W1   (M=32768,K=512,N=512) ----------------
// Block = 256 thr (8 waves); A tile async-staged to LDS once, each wave owns a
// 16x64 output strip (4 WMMA frags sharing one A fragment per k-step).
__global__ __launch_bounds__(256) void gemm1_kernel(const float* __restrict__ X,
                                                    float* __restrict__ ws) {
  __shared__ float As[16 * LDSP];
  const float* W1 = ws + WS_W1;
  float* buf = ws + WS_BUF;
  int tid  = threadIdx.x;
  int lane = tid & 31;
  int wave = tid >> 5;
  int m0 = blockIdx.x * 16;
  int n0 = wave * 64;

  stage_tile_async(X + (size_t)m0 * 512, As, tid);
  wait_async_all();
  __syncthreads();

  v8f c0 = {}, c1 = {}, c2 = {}, c3 = {};
#pragma unroll 2
  for (int k = 0; k < D_MODEL; k += 4) {
    v2f a  = load_a_lds(As, k, lane);
    v2f b0 = load_b_pair(W1, k, n0 + 0,  lane);
    v2f b1 = load_b_pair(W1, k, n0 + 16, lane);
    v2f b2 = load_b_pair(W1, k, n0 + 32, lane);
    v2f b3 = load_b_pair(W1, k, n0 + 48, lane);
    c0 = wmma_f32(a, b0, c0);
    c1 = wmma_f32(a, b1, c1);
    c2 = wmma_f32(a, b2, c2);
    c3 = wmma_f32(a, b3, c3);
  }
  store_c(buf, 512, m0, n0 + 0,  lane, c0);
  store_c(buf, 512, m0, n0 + 16, lane, c1);
  store_c(buf, 512, m0, n0 + 32, lane, c2);
  store_c(buf, 512, m0, n0 + 48, lane, c3);
}

// ---------------- scan: in-place complex recurrence over t<len[b] -----------
__global__ __launch_bounds__(32) void scan_kernel(
    const long long* __restrict__ lengths, float* __restrict__ ws) {
  int tid = blockIdx.x * blockDim.x + threadIdx.x;  // 4096 = 16*256
  int n = tid & 255;
  int b = tid >> 8;
  if (b >= BATCH) return;
  const float* ab = ws + WS_ABAR;
  float are = ab[n], aim = ab[256 + n];
  float* base = ws + WS_BUF + (size_t)b * SEQ * 512 + n;
  int len = (int)lengths[b];
  float sre = 0.0f, sim = 0.0f;
#pragma unroll 4
  for (int t = 0; t < len; ++t) {
    float* p = base + (size_t)t * 512;
    __builtin_prefetch(p + 16 * 512, 0, 0);  // speculative stream-ahead
    float ire = p[0];
    float iim = p[256];
    float nre = fmaf(sre, are, fmaf(-sim, aim, ire));
    float nim = fmaf(sre, aim, fmaf(sim, are, iim));
    sre = nre;
    sim = nim;
    p[0]   = sre;
    p[256] = sim;
  }
}

// ------- GEMM2: y = [st_re|st_im]@W2[0:512] + x@W2[512:1024], length-masked --
__global__ __launch_bounds__(256) void gemm2_kernel(
    const float* __restrict__ X, const long long* __restrict__ lengths,
    const float* __restrict__ ws, float* __restrict__ Y) {
  __shared__ float As[2 * 16 * LDSP];
  const float* W2  = ws + WS_W2;
  const float* buf = ws + WS_BUF;
  int tid  = threadIdx.x;
  int lane = tid & 31;
  int wave = tid >> 5;
  int m0 = blockIdx.x * 16;
  int n0 = wave * 64;
  int b  = m0 / SEQ;  // SEQ % 16 == 0 -> whole tile same batch
  int l0 = m0 % SEQ;
  long long len = lengths[b];

  v8f c0 = {}, c1 = {}, c2 = {}, c3 = {};
  if ((long long)l0 < len) {  // uniform per block; all-invalid tiles store 0
    stage_tile_async(buf + (size_t)m0 * 512, As, tid);                // states
    stage_tile_async(X   + (size_t)m0 * 512, As + 16 * LDSP, tid);    // x
    wait_async_all();
    __syncthreads();

    const float* W2d = W2 + (size_t)256 * 1024;  // rows 512..1023 (pairs)
#pragma unroll 2
    for (int k = 0; k < 512; k += 4) {  // states segment
      v2f a  = load_a_lds(As, k, lane);
      v2f b0 = load_b_pair(W2, k, n0 + 0,  lane);
      v2f b1 = load_b_pair(W2, k, n0 + 16, lane);
      v2f b2 = load_b_pair(W2, k, n0 + 32, lane);
      v2f b3 = load_b_pair(W2, k, n0 + 48, lane);
      c0 = wmma_f32(a, b0, c0);
      c1 = wmma_f32(a, b1, c1);
      c2 = wmma_f32(a, b2, c2);
      c3 = wmma_f32(a, b3, c3);
    }
#pragma unroll 2
    for (int k = 0; k < 512; k += 4) {  // D-residual segment
      v2f a  = load_a_lds(As + 16 * LDSP, k, lane);
      v2f b0 = load_b_pair(W2d, k, n0 + 0,  lane);
      v2f b1 = load_b_pair(W2d, k, n0 + 16, lane);
      v2f b2 = load_b_pair(W2d, k, n0 + 32, lane);
      v2f b3 = load_b_pair(W2d, k, n0 + 48, lane);
      c0 = wmma_f32(a, b0, c0);
      c1 = wmma_f32(a, b1, c1);
      c2 = wmma_f32(a, b2, c2);
      c3 = wmma_f32(a, b3, c3);
    }
  }
  // masked store
  int col = lane & 15;
  int mb  = (lane >> 4) << 3;
  v8f frags[4] = {c0, c1, c2, c3};
#pragma unroll
  for (int f = 0; f < 4; ++f) {
#pragma unroll
    for (int i = 0; i < 8; ++i) {
      int l = l0 + mb + i;
      float v = ((long long)l < len) ? frags[f][i] : 0.0f;
      Y[(size_t)(m0 + mb + i) * D_MODEL + n0 + f * 16 + col] = v;
    }
  }
}

extern "C" void kernel_launch(void* const* d_in, const int* in_sizes, int n_in,
                              void* d_out, int out_size, void* d_ws, size_t ws_size,
                              hipStream_t stream) {
  (void)in_sizes; (void)n_in; (void)out_size; (void)ws_size;
  const float*     x       = (const float*)d_in[0];
  const long long* lengths = (const long long*)d_in[1];
  const float*     lrl     = (const float*)d_in[2];
  const float*     limg    = (const float*)d_in[3];
  const float*     logdt   = (const float*)d_in[4];
  const float*     Bre     = (const float*)d_in[5];
  const float*     Bim     = (const float*)d_in[6];
  const float*     Cre     = (const float*)d_in[7];
  const float*     Cim     = (const float*)d_in[8];
  const float*     Dw      = (const float*)d_in[9];
  float* ws  = (float*)d_ws;
  float* out = (float*)d_out;

  setup_w1_kernel<<<(512 * 256) / 256, 256, 0, stream>>>(lrl, limg, logdt, Bre, Bim, ws);
  setup_w2_kernel<<<(1024 * 512) / 256, 256, 0, stream>>>(Cre, Cim, Dw, ws);
  gemm1_kernel<<<M_TOTAL / 16, 256, 0, stream>>>(x, ws);
  scan_kernel<<<128, 32, 0, stream>>>(lengths, ws);
  gemm2_kernel<<<M_TOTAL / 16, 256, 0, stream>>>(x, lengths, ws, out);
}